// MeanMessageAggregator_64707977282176
// MI455X (gfx1250) — compile-verified
//
#include <hip/hip_runtime.h>
#include <hip/hip_bf16.h>
#include <stdint.h>

// MeanMessageAggregator for MI455X (gfx1250).
// Segment mean + segment max: E=1M messages x D=256 f32 -> N=64K nodes.
// ~1.1 GB HBM traffic @ 23.3 TB/s => ~50us floor; 0.25 flop/byte so WMMA is
// inapplicable. CDNA5 features used: async global->LDS DMA double-buffered on
// ASYNCcnt (global_load_async_to_lds_b128 + s_wait_asynccnt), conflict-free
// ds_load_b128 consume, and fire-and-forget global_atomic_add_f32 RMWs that
// resolve in the 192 MB L2 (the 64 MB accumulator is L2-resident).

#define D_FEAT          256
#define TILE_MSGS       32                       // messages per tile
#define TILES_PER_BLOCK 8
#define BLOCK           256                      // 8 waves (wave32)
#define TILE_F4         (TILE_MSGS * D_FEAT / 4) // 2048 float4 = 32 KB
#define F4_PER_THREAD   (TILE_F4 / BLOCK)        // 8 async b128 ops per wave/tile

// ---------------------------------------------------------------- init (float4)
__global__ void agg_init_kernel(float* __restrict__ out,
                                unsigned* __restrict__ counts,
                                int N, int D) {
    long long i4 = (long long)blockIdx.x * blockDim.x + threadIdx.x;
    long long nd4 = (long long)N * D / 4;
    if (i4 < nd4) {
        float4 z = {0.f, 0.f, 0.f, 0.f};
        ((float4*)(out + N))[i4] = z;            // sum accumulators
    }
    if (i4 < N) {
        out[i4] = (float)i4;                     // node_ids = arange(N)
        out[N + (long long)N * D + i4] = 0.0f;   // ts max accum (timestamps >= 0)
        counts[i4] = 0u;
    }
}

// ---------------------------------------------------------------- scatter
__global__ void __launch_bounds__(BLOCK)
agg_scatter_kernel(const float* __restrict__ messages,
                   const float* __restrict__ timestamp,
                   const int*   __restrict__ node_index,
                   float* __restrict__ out,
                   unsigned* __restrict__ counts,
                   int E, int N) {
    __shared__ float4 smem[2][TILE_F4];          // 64 KB of the 320 KB WGP LDS

    const int tid  = threadIdx.x;
    const int w    = tid >> 5;                   // wave id 0..7
    const int lane = tid & 31;
    float* out_sums = out + N;
    float* out_ts   = out + N + (size_t)N * D_FEAT;

    const int ntiles = E / TILE_MSGS;
    const int tile0  = blockIdx.x * TILES_PER_BLOCK;

    // Issue one 32 KB tile as async global->LDS b128 copies: 8 waves x 8 ops,
    // each op moves 32 lanes x 16B = 512B contiguous -> fully coalesced.
    auto issue_tile = [&](int buf, int tile) {
        int t = tile < ntiles ? tile : (ntiles - 1);              // clamp, never OOB
        const float4* gsrc = (const float4*)messages +
                             (size_t)t * TILE_MSGS * (D_FEAT / 4);
        unsigned lds_base = (unsigned)(uintptr_t)(&smem[buf][0]); // low 32b of flat = LDS offset
        #pragma unroll
        for (int j = 0; j < F4_PER_THREAD; ++j) {
            int e4 = j * BLOCK + tid;                             // float4 index in tile
            unsigned lds_off = lds_base + (unsigned)e4 * 16u;
            const float4* gp = gsrc + e4;
            asm volatile("global_load_async_to_lds_b128 %0, %1, off"
                         :: "v"(lds_off), "v"(gp) : "memory");
        }
    };

    issue_tile(0, tile0);

    for (int i = 0; i < TILES_PER_BLOCK; ++i) {
        const int tile = tile0 + i;
        const int cur  = i & 1;

        if (i + 1 < TILES_PER_BLOCK) {
            issue_tile(cur ^ 1, tile + 1);
            // prev tile's 8 ops done (in-order per counter); next 8 still in flight
            asm volatile("s_wait_asynccnt 8" ::: "memory");
        } else {
            asm volatile("s_wait_asynccnt 0" ::: "memory");
        }
        __syncthreads();

        if (tile < ntiles) {
            // Wave w owns messages 4w..4w+3 of the tile. Hoist their node ids
            // (uniform across the wave -> scalarized by the compiler).
            const int ebase = tile * TILE_MSGS + 4 * w;
            int nd[4];
            #pragma unroll
            for (int m = 0; m < 4; ++m) nd[m] = node_index[ebase + m];

            // Conflict-free consume: on step j all 32 lanes read message
            // 4w+(j>>1), half-row (j&1)*32+lane -> each ds_load_b128 covers
            // 512B contiguous (all 64 banks), and the 4 atomics per lane form
            // a contiguous 512B burst into one node row in L2.
            #pragma unroll
            for (int j = 0; j < 8; ++j) {
                const int mloc = j >> 1;                        // compile-time per iter
                const int col4 = (j & 1) * 32 + lane;           // float4 column 0..63
                float4 v = smem[cur][(4 * w + mloc) * (D_FEAT / 4) + col4];
                float* p = out_sums + (size_t)nd[mloc] * D_FEAT + (size_t)col4 * 4;
                asm volatile("global_atomic_add_f32 %0, %1, off" :: "v"(p    ), "v"(v.x) : "memory");
                asm volatile("global_atomic_add_f32 %0, %1, off" :: "v"(p + 1), "v"(v.y) : "memory");
                asm volatile("global_atomic_add_f32 %0, %1, off" :: "v"(p + 2), "v"(v.z) : "memory");
                asm volatile("global_atomic_add_f32 %0, %1, off" :: "v"(p + 3), "v"(v.w) : "memory");
            }

            if (lane < 4) {                                     // one lane per message
                const int e = ebase + lane;
                atomicAdd(&counts[node_index[e]], 1u);
                // timestamps >= 0, so unsigned bit-pattern max == float max
                atomicMax((unsigned*)(out_ts + node_index[e]),
                          __float_as_uint(timestamp[e]));
            }
        }
        __syncthreads();  // buffer `cur` is re-issued next iteration
    }
}

// ---------------------------------------------------------------- finalize (float4)
__global__ void agg_finalize_kernel(float* __restrict__ out,
                                    const unsigned* __restrict__ counts,
                                    int N, int D) {
    long long i4 = (long long)blockIdx.x * blockDim.x + threadIdx.x;
    long long nd4 = (long long)N * D / 4;
    if (i4 < nd4) {
        unsigned c = counts[i4 / (D / 4)];
        if (c == 0u) c = 1u;                     // every node appears, but be safe
        float r = 1.0f / (float)c;
        float4* p = (float4*)(out + N) + i4;     // out+N is 256 KB-aligned
        float4 v = *p;
        v.x *= r; v.y *= r; v.z *= r; v.w *= r;
        *p = v;
    }
}

// ---------------------------------------------------------------- launch
extern "C" void kernel_launch(void* const* d_in, const int* in_sizes, int n_in,
                              void* d_out, int out_size, void* d_ws, size_t ws_size,
                              hipStream_t stream) {
    const float* messages   = (const float*)d_in[0];   // [E, D] f32
    const float* timestamp  = (const float*)d_in[1];   // [E]    f32
    const int*   node_index = (const int*)d_in[2];     // [E]    i32
    // d_in[3] = num_nodes (device scalar; N recovered from out_size instead)

    const int E = in_sizes[1];
    const int D = in_sizes[0] / E;                     // 256
    const int N = out_size / (D + 2);                  // out = [N ids | N*D means | N ts]

    float*    out    = (float*)d_out;
    unsigned* counts = (unsigned*)d_ws;                // N * 4B = 256 KB scratch

    const long long nd4 = (long long)N * D / 4;
    const int vec_blocks = (int)((nd4 + BLOCK - 1) / BLOCK);
    agg_init_kernel<<<vec_blocks, BLOCK, 0, stream>>>(out, counts, N, D);

    const int msgs_per_block = TILE_MSGS * TILES_PER_BLOCK;            // 256
    const int scat_blocks = (E + msgs_per_block - 1) / msgs_per_block; // 4096
    agg_scatter_kernel<<<scat_blocks, BLOCK, 0, stream>>>(
        messages, timestamp, node_index, out, counts, E, N);

    agg_finalize_kernel<<<vec_blocks, BLOCK, 0, stream>>>(out, counts, N, D);
}